// VoxelRasterizer_52974126629402
// MI455X (gfx1250) — compile-verified
//
#include <hip/hip_runtime.h>
#include <hip/hip_bf16.h>
#include <stdint.h>

static constexpr int kH = 1024;
static constexpr int kW = 1024;
static constexpr int kHW = kH * kW;
static constexpr float kAlphaThr = 0.95f;

typedef float v2f __attribute__((ext_vector_type(2)));
typedef float v8f __attribute__((ext_vector_type(8)));

struct Node { int next; float depth; float a; float r; float g; float b; };

// ---------------------------------------------------------------------------
// Kernel 1: build MVP (proj @ view) from camera params. One thread.
// ---------------------------------------------------------------------------
__global__ void mvp_kernel(const float* __restrict__ cam_pos,
                           const float* __restrict__ cam_fwd,
                           const float* __restrict__ cam_up,
                           float* __restrict__ mvp)
{
    if (threadIdx.x != 0 || blockIdx.x != 0) return;
    float fx = cam_fwd[0], fy = cam_fwd[1], fz = cam_fwd[2];
    float il = rsqrtf(fx*fx + fy*fy + fz*fz);
    fx *= il; fy *= il; fz *= il;
    float ux = cam_up[0], uy = cam_up[1], uz = cam_up[2];
    float rx = fy*uz - fz*uy, ry = fz*ux - fx*uz, rz = fx*uy - fy*ux;
    il = rsqrtf(rx*rx + ry*ry + rz*rz);
    rx *= il; ry *= il; rz *= il;
    float vx = ry*fz - rz*fy, vy = rz*fx - rx*fz, vz = rx*fy - ry*fx;
    float px = cam_pos[0], py = cam_pos[1], pz = cam_pos[2];

    float view[4][4] = {};
    view[0][0]=rx;  view[0][1]=ry;  view[0][2]=rz;  view[0][3]=-(rx*px+ry*py+rz*pz);
    view[1][0]=vx;  view[1][1]=vy;  view[1][2]=vz;  view[1][3]=-(vx*px+vy*py+vz*pz);
    view[2][0]=-fx; view[2][1]=-fy; view[2][2]=-fz; view[2][3]= (fx*px+fy*py+fz*pz);
    view[3][3]=1.0f;

    const float NEARP = 0.1f, FARP = 500.0f;
    const float ff = 1.7320508075688772f;   // 1 / tan(60deg / 2)
    float proj[4][4] = {};
    proj[0][0]=ff; proj[1][1]=ff;
    proj[2][2]=(FARP+NEARP)/(NEARP-FARP);
    proj[2][3]=2.0f*FARP*NEARP/(NEARP-FARP);
    proj[3][2]=-1.0f;

    for (int i = 0; i < 4; ++i)
        for (int j = 0; j < 4; ++j) {
            float s = 0.f;
            for (int k = 0; k < 4; ++k) s += proj[i][k]*view[k][j];
            mvp[i*4+j] = s;
        }
}

// ---------------------------------------------------------------------------
// Kernel 2: reset per-pixel fragment-list heads.
// ---------------------------------------------------------------------------
__global__ void init_heads(int* __restrict__ heads)
{
    int i = blockIdx.x * blockDim.x + threadIdx.x;
    if (i < kHW) heads[i] = -1;
}

// ---------------------------------------------------------------------------
// Kernel 3: async-stage coords to LDS, transform 32 voxels/wave with two
// V_WMMA_F32_16X16X4_F32 ops, scatter fragments into per-pixel linked lists.
// ---------------------------------------------------------------------------
__global__ void __launch_bounds__(256)
transform_kernel(const float* __restrict__ coords,
                 const float* __restrict__ feats,
                 const float* __restrict__ mvp,
                 int* __restrict__ heads,
                 Node* __restrict__ nodes,
                 int n)
{
    __shared__ float lclip[8][16][32];  // [wave][clip component][voxel-in-wave]
    __shared__ float lcoord[8][96];     // [wave][32 voxels * xyz]

    const int lane = threadIdx.x & 31;
    const int wv   = threadIdx.x >> 5;
    const int wave = (blockIdx.x * blockDim.x + threadIdx.x) >> 5;
    const int base = wave * 32;
    if (base >= n) return;              // wave-uniform exit, keeps EXEC all-1s

    // ---- Async copy: 96 contiguous floats (32 voxels * xyz) -> LDS.
    // One element per lane per op; 3 ops cover the tile. Tracked by ASYNCcnt.
    const int n3 = 3 * n;
#pragma unroll
    for (int k = 0; k < 3; ++k) {
        int gi = 3 * base + k * 32 + lane;
        gi = (gi < n3) ? gi : (n3 - 1);
        unsigned lo = (unsigned)(size_t)&lcoord[wv][k * 32 + lane];
        unsigned long long gp = (unsigned long long)(coords + gi);
        asm volatile("global_load_async_to_lds_b32 %0, %1, off"
                     :: "v"(lo), "v"(gp) : "memory");
    }

    // ---- B matrix (4x16 f32): B[kk][col] = mvp[col][kk] (MVP^T), cols 4..15 = 0.
    // wave32 layout: lanes 0-15 hold K=0,1 (vgpr0,vgpr1); lanes 16-31 hold K=2,3.
    // Unconditional clamped loads + cndmask select (no predicated load regions).
    const int  ncol  = lane & 15;
    const int  half  = lane >> 4;
    const bool colok = ncol < 4;
    const int  mrow  = ncol & 3;
    float b0 = mvp[mrow * 4 + 2 * half    ];
    float b1 = mvp[mrow * 4 + 2 * half + 1];
    v2f bmat;
    bmat.x = colok ? b0 : 0.0f;
    bmat.y = colok ? b1 : 0.0f;

    asm volatile("s_wait_asynccnt 0x0" ::: "memory");

    // ---- A matrices (16x4 f32) from the LDS-staged tile.
    // lanes 0-15: K=0,1 -> (x,y); lanes 16-31: K=2,3 -> (z,1).
    int l1 = ((base + ncol      < n) ? (base + ncol)      : (n - 1)) - base;
    int l2 = ((base + 16 + ncol < n) ? (base + 16 + ncol) : (n - 1)) - base;
    float c1x = lcoord[wv][3*l1], c1y = lcoord[wv][3*l1+1], c1z = lcoord[wv][3*l1+2];
    float c2x = lcoord[wv][3*l2], c2y = lcoord[wv][3*l2+1], c2z = lcoord[wv][3*l2+2];
    v2f a1, a2;
    a1.x = half ? c1z : c1x;  a1.y = half ? 1.0f : c1y;
    a2.x = half ? c2z : c2x;  a2.y = half ? 1.0f : c2y;

    v8f cz = {};
    // 8 args: (neg_a, A, neg_b, B, c_mod, C, reuse_a, reuse_b)
    v8f d1 = __builtin_amdgcn_wmma_f32_16x16x4_f32(false, a1, false, bmat,
                                                   (short)0, cz, false, false);
    v8f d2 = __builtin_amdgcn_wmma_f32_16x16x4_f32(false, a2, false, bmat,
                                                   (short)0, cz, false, false);

    // ---- D layout (16x16 f32): element (m, c) lives in vgpr m%8, lane c + 16*(m/8).
    // Writer lane (col c = 0..3 on each half) stores its 8 contiguous D values
    // component-major, so the stores merge into ds_store_b128s.
    if (colok) {
        float* row = &lclip[wv][ncol][half * 8];
#pragma unroll
        for (int r = 0; r < 8; ++r) {
            row[r]      = d1[r];   // voxels  0..15
            row[16 + r] = d2[r];   // voxels 16..31
        }
    }
    asm volatile("s_wait_dscnt 0x0" ::: "memory");   // wave-local LDS RAW fence

    float cx = lclip[wv][0][lane];
    float cy = lclip[wv][1][lane];
    float cc = lclip[wv][2][lane];
    float cw = lclip[wv][3][lane];

    float iw  = 1.0f / (cw + 1e-8f);
    float nx  = cx * iw, ny = cy * iw, nz = cc * iw;
    bool  vis = (nx >= -1.f) & (nx <= 1.f) &
                (ny >= -1.f) & (ny <= 1.f) &
                (nz >= -1.f) & (nz <= 1.f);
    float sxf = floorf((nx + 1.0f) * 0.5f * (float)kW);
    sxf = fminf(fmaxf(sxf, 0.0f), (float)(kW - 1));
    float syf = floorf((1.0f - ny) * 0.5f * (float)kH);
    syf = fminf(fmaxf(syf, 0.0f), (float)(kH - 1));
    int pix = (int)syf * kW + (int)sxf;

    int i = base + lane;
    if (i < n) {
        float a = vis ? feats[7*i + 5] : 0.0f;
        a = fminf(fmaxf(a, 0.0f), 1.0f - 1e-6f);
        Node nd;
        nd.depth = nz;
        nd.a     = a;
        nd.r     = feats[7*i + 0];
        nd.g     = feats[7*i + 1];
        nd.b     = feats[7*i + 2];
        nd.next  = atomicExch(&heads[pix], i);
        nodes[i] = nd;
    }
}

// ---------------------------------------------------------------------------
// Kernel 4: per-pixel ordered resolve. Total order = (depth desc, index asc),
// matching lexsort((-depth, pix)) with stable tie-break. Mean list length ~2;
// the whole node pool sits in the 192MB L2, so repeated list walks are cheap.
// ---------------------------------------------------------------------------
__device__ __forceinline__ unsigned ordf(float f)
{
    unsigned u = __float_as_uint(f);
    return (u & 0x80000000u) ? ~u : (u | 0x80000000u);   // monotone float->uint
}

__global__ void resolve_kernel(const int* __restrict__ heads,
                               const Node* __restrict__ nodes,
                               float* __restrict__ out)
{
    int pix = blockIdx.x * blockDim.x + threadIdx.x;
    if (pix >= kHW) return;
    int head = heads[pix];

    float Dr = 0.f, Dg = 0.f, Db = 0.f;
    float sumL = 0.f, sumL2 = 0.f;
    unsigned long long last = 0ull;
    bool first = true;

    for (;;) {
        unsigned long long best = ~0ull;
        float ba = 0.f, br = 0.f, bg = 0.f, bb = 0.f;
        int found = 0;
        for (int j = head; j >= 0; j = nodes[j].next) {
            unsigned long long rk =
                ((unsigned long long)(~ordf(nodes[j].depth)) << 32) | (unsigned)j;
            if ((first || rk > last) && rk < best) {
                best = rk;
                ba = nodes[j].a; br = nodes[j].r; bg = nodes[j].g; bb = nodes[j].b;
                found = 1;
            }
        }
        if (!found) break;
        last = best; first = false;

        float alpha_before = 1.0f - __expf(sumL);
        float a2 = (alpha_before < kAlphaThr) ? ba : 0.0f;
        sumL += log1pf(-ba);
        float om = 1.0f - a2;
        Dr = Dr * om + br * a2;
        Dg = Dg * om + bg * a2;
        Db = Db * om + bb * a2;
        sumL2 += log1pf(-a2);
    }

    float A   = 1.0f - __expf(sumL2);
    float inv = (A > 0.0f) ? (1.0f / A) : 0.0f;
    out[0*kHW + pix] = Dr * inv;
    out[1*kHW + pix] = Dg * inv;
    out[2*kHW + pix] = Db * inv;
}

// ---------------------------------------------------------------------------
extern "C" void kernel_launch(void* const* d_in, const int* in_sizes, int n_in,
                              void* d_out, int out_size, void* d_ws, size_t ws_size,
                              hipStream_t stream)
{
    const float* coords  = (const float*)d_in[0];
    const float* feats   = (const float*)d_in[1];
    const float* cam_pos = (const float*)d_in[2];
    const float* cam_fwd = (const float*)d_in[3];
    const float* cam_up  = (const float*)d_in[4];
    float* out = (float*)d_out;
    int n = in_sizes[0] / 3;

    char*  ws    = (char*)d_ws;
    float* mvp   = (float*)ws;                                    //    64 B
    int*   heads = (int*)(ws + 256);                              //  4 MB
    Node*  nodes = (Node*)(ws + 256 + (size_t)kHW * sizeof(int)); // 48 MB

    mvp_kernel<<<1, 1, 0, stream>>>(cam_pos, cam_fwd, cam_up, mvp);
    init_heads<<<(kHW + 255) / 256, 256, 0, stream>>>(heads);
    transform_kernel<<<(n + 255) / 256, 256, 0, stream>>>(coords, feats, mvp,
                                                          heads, nodes, n);
    resolve_kernel<<<(kHW + 255) / 256, 256, 0, stream>>>(heads, nodes, out);
}